// Net_15865609192050
// MI455X (gfx1250) — compile-verified
//
#include <hip/hip_runtime.h>
#include <hip/hip_bf16.h>
#include <math.h>

// Problem constants (match reference)
constexpr int B_  = 128;
constexpr int NP_ = 2048;
constexpr int F0_ = 64;
constexpr int H_  = 32;
constexpr int C_  = 8;
constexpr int DEG_ = 16;
constexpr int NT_ = B_ * NP_;        // 262144
constexpr int EP_ = NP_ * DEG_;      // 32768 edges per graph
constexpr int K1_ = 615, K2_ = 185, K3_ = 56;
constexpr float NEG_ = -1e30f;

typedef float v2f __attribute__((ext_vector_type(2)));
typedef float v8f __attribute__((ext_vector_type(8)));

// ---------------------------------------------------------------------------
// Segment-sum (GraphConv aggregation) via LDS-staged atomics.
// One block per (graph, 8-feature chunk). 2048 nodes x 8 feats = 64KB LDS.
// ---------------------------------------------------------------------------
template <int F>
__global__ __launch_bounds__(512) void agg_kernel(
    const float* __restrict__ x, const int* __restrict__ esrc,
    const int* __restrict__ edst, float* __restrict__ agg)
{
    constexpr int CH = 8;
    __shared__ float sh[NP_ * CH];            // 65536 bytes
    const int g   = blockIdx.x;
    const int c0  = blockIdx.y * CH;
    const int tid = threadIdx.x;

    for (int i = tid; i < NP_ * CH; i += 512) sh[i] = 0.0f;
    __syncthreads();

    const size_t eb = (size_t)g * EP_;
    const int nodeBase = g * NP_;
    for (int e = tid; e < EP_; e += 512) {
        const int s = esrc[eb + e];           // absolute src node
        const int d = edst[eb + e] - nodeBase;// local dst node (0..2047)
        const float* xp = x + (size_t)s * F + c0;
        const float4 v0 = *(const float4*)(xp);
        const float4 v1 = *(const float4*)(xp + 4);
        float* dp = sh + d * CH;
        atomicAdd(dp + 0, v0.x); atomicAdd(dp + 1, v0.y);
        atomicAdd(dp + 2, v0.z); atomicAdd(dp + 3, v0.w);
        atomicAdd(dp + 4, v1.x); atomicAdd(dp + 5, v1.y);
        atomicAdd(dp + 6, v1.z); atomicAdd(dp + 7, v1.w);
    }
    __syncthreads();

    for (int i = tid; i < NP_ * CH; i += 512) {
        const int n = i >> 3, f = i & 7;
        agg[(size_t)(nodeBase + n) * F + c0 + f] = sh[i];
    }
}

// ---------------------------------------------------------------------------
// GraphConv: y = relu(x @ Wr + agg @ Wn + b), N = 32 output features.
// Wave32 WMMA f32 16x16x4. Each wave: one 16-row M tile, two v8f accs (N=32).
// A layout (16x4 f32): lane<16 -> M=lane, VGPR0/1 = K {k0,k0+1};
//                      lane>=16 -> M=lane-16, K {k0+2,k0+3}  => float2 load.
// B layout (4x16):     VGPR v, half h -> K = k0 + v + 2h, N = lane&15.
// C/D layout: VGPR r -> M = r + 8*(lane>=16), N = lane&15.
// ---------------------------------------------------------------------------
template <int K>
__global__ __launch_bounds__(256) void conv_kernel(
    const float* __restrict__ xin, const float* __restrict__ agg,
    const float* __restrict__ Wr, const float* __restrict__ Wn,
    const float* __restrict__ bias, float* __restrict__ y)
{
    const int lane = threadIdx.x & 31;
    const int wave = threadIdx.x >> 5;
    const int m0   = (blockIdx.x * 8 + wave) * 16;
    const int hi   = lane >> 4;               // 0 or 1
    const int l16  = lane & 15;
    const int row  = m0 + l16;

    v8f acc0 = {};
    v8f acc1 = {};

    const float* A = xin;
    const float* W = Wr;
    #pragma unroll 1
    for (int pass = 0; pass < 2; ++pass) {
        #pragma unroll
        for (int k0 = 0; k0 < K; k0 += 4) {
            const int kk = k0 + 2 * hi;
            v2f a = *(const v2f*)(A + (size_t)row * K + kk);   // b64 load
            v2f b0, b1;
            b0.x = W[(kk    ) * H_ + l16];
            b0.y = W[(kk + 1) * H_ + l16];
            b1.x = W[(kk    ) * H_ + 16 + l16];
            b1.y = W[(kk + 1) * H_ + 16 + l16];
            acc0 = __builtin_amdgcn_wmma_f32_16x16x4_f32(
                false, a, false, b0, (short)0, acc0, false, false);
            acc1 = __builtin_amdgcn_wmma_f32_16x16x4_f32(
                false, a, false, b1, (short)0, acc1, false, false);
        }
        A = agg; W = Wn;
    }

    const float bc0 = bias[l16];
    const float bc1 = bias[16 + l16];
    #pragma unroll
    for (int r = 0; r < 8; ++r) {
        const int orow = m0 + hi * 8 + r;
        y[(size_t)orow * H_ + l16]      = fmaxf(acc0[r] + bc0, 0.0f);
        y[(size_t)orow * H_ + 16 + l16] = fmaxf(acc1[r] + bc1, 0.0f);
    }
}

// ---------------------------------------------------------------------------
// TopKPooling + fused readout. One block (256 thr) per graph.
// score = (x.p)/||p|| masked by prev mask; LDS bitonic sort (desc, idx asc on
// ties == jax.lax.top_k); x <- x*tanh(score)*kept (in place); mask <- kept;
// r[g, 0:32] += masked-max, r[g, 32:64] += masked-mean.
// ---------------------------------------------------------------------------
__global__ __launch_bounds__(256) void pool_kernel(
    float* __restrict__ x, unsigned char* __restrict__ mask,
    const float* __restrict__ p, int Ksel, float* __restrict__ r)
{
    __shared__ float sc[NP_];
    __shared__ int   id[NP_];
    __shared__ float tn[NP_];
    __shared__ unsigned char kept[NP_];
    __shared__ float sp[H_];
    __shared__ float redmax[256];
    __shared__ float redsum[256];

    const int tid = threadIdx.x;
    const int g   = blockIdx.x;
    const size_t gbase = (size_t)g * NP_;

    if (tid < H_) sp[tid] = p[tid];
    __syncthreads();
    float pn2 = 0.0f;
    #pragma unroll
    for (int f = 0; f < H_; ++f) pn2 += sp[f] * sp[f];
    const float inv_pn = 1.0f / sqrtf(pn2);

    // scores
    for (int n = tid; n < NP_; n += 256) {
        const float* xr = x + (gbase + n) * H_;
        float d = 0.0f;
        #pragma unroll
        for (int f = 0; f < H_; ++f) d += xr[f] * sp[f];
        const float s = mask[gbase + n] ? d * inv_pn : NEG_;
        sc[n] = s; tn[n] = s; id[n] = n;
    }
    __syncthreads();

    // bitonic sort: descending score, ascending index on ties
    for (int kk = 2; kk <= NP_; kk <<= 1) {
        for (int j = kk >> 1; j > 0; j >>= 1) {
            for (int i = tid; i < NP_; i += 256) {
                const int ixj = i ^ j;
                if (ixj > i) {
                    const float s1 = sc[i], s2 = sc[ixj];
                    const int   i1 = id[i], i2 = id[ixj];
                    const bool firstGreater = (s1 > s2) || (s1 == s2 && i1 < i2);
                    const bool largerFirst  = ((i & kk) == 0);
                    if (firstGreater != largerFirst) {
                        sc[i] = s2; sc[ixj] = s1;
                        id[i] = i2; id[ixj] = i1;
                    }
                }
            }
            __syncthreads();
        }
    }

    // kept flags + tanh(score)
    for (int n = tid; n < NP_; n += 256) kept[n] = 0;
    __syncthreads();
    for (int t = tid; t < Ksel; t += 256) kept[id[t]] = 1;
    __syncthreads();
    for (int n = tid; n < NP_; n += 256) tn[n] = tanhf(tn[n]);
    __syncthreads();

    // x update + masked max/mean readout. thread = (feature f, node-slice sub)
    const int f = tid & 31, sub = tid >> 5;    // 32 feats x 8 slices
    float lmax = NEG_, lsum = 0.0f;
    for (int n = sub; n < NP_; n += 8) {
        float* xp = x + (gbase + n) * H_ + f;
        float v = 0.0f;
        if (kept[n]) {
            v = (*xp) * tn[n];
            lmax = fmaxf(lmax, v);
            lsum += v;
        }
        *xp = v;
    }
    redmax[tid] = lmax;
    redsum[tid] = lsum;
    for (int n = tid; n < NP_; n += 256) mask[gbase + n] = kept[n];
    __syncthreads();

    if (sub == 0) {
        float m = redmax[f], s = redsum[f];
        #pragma unroll
        for (int ss = 1; ss < 8; ++ss) {
            m = fmaxf(m, redmax[ss * 32 + f]);
            s += redsum[ss * 32 + f];
        }
        r[g * 64 + f]       += m;
        r[g * 64 + 32 + f]  += s / (float)Ksel;
    }
}

// ---------------------------------------------------------------------------
// Head: h = relu(r @ Wl1 + bl1); out = log_softmax(h @ Wl3 + bl3)
// One thread per graph (128 threads, 1 block) — tiny.
// ---------------------------------------------------------------------------
__global__ __launch_bounds__(128) void head_kernel(
    const float* __restrict__ r, const float* __restrict__ Wl1,
    const float* __restrict__ bl1, const float* __restrict__ Wl3,
    const float* __restrict__ bl3, float* __restrict__ out)
{
    const int g = threadIdx.x;
    float h[H_];
    #pragma unroll
    for (int j = 0; j < H_; ++j) {
        float s = bl1[j];
        #pragma unroll
        for (int i = 0; i < 2 * H_; ++i) s += r[g * 64 + i] * Wl1[i * H_ + j];
        h[j] = fmaxf(s, 0.0f);
    }
    float o[C_];
    float mx = NEG_;
    #pragma unroll
    for (int c = 0; c < C_; ++c) {
        float s = bl3[c];
        #pragma unroll
        for (int j = 0; j < H_; ++j) s += h[j] * Wl3[j * C_ + c];
        o[c] = s;
        mx = fmaxf(mx, s);
    }
    float se = 0.0f;
    #pragma unroll
    for (int c = 0; c < C_; ++c) se += expf(o[c] - mx);
    const float lse = mx + logf(se);
    #pragma unroll
    for (int c = 0; c < C_; ++c) out[g * C_ + c] = o[c] - lse;
}

// ---------------------------------------------------------------------------
extern "C" void kernel_launch(void* const* d_in, const int* in_sizes, int n_in,
                              void* d_out, int out_size, void* d_ws, size_t ws_size,
                              hipStream_t stream)
{
    (void)in_sizes; (void)n_in; (void)out_size; (void)ws_size;

    const float* x    = (const float*)d_in[0];
    const int*   esrc = (const int*)d_in[1];
    const int*   edst = (const int*)d_in[2];
    const float* Wr1  = (const float*)d_in[3];
    const float* Wn1  = (const float*)d_in[4];
    const float* b1   = (const float*)d_in[5];
    const float* Wr2  = (const float*)d_in[6];
    const float* Wn2  = (const float*)d_in[7];
    const float* b2   = (const float*)d_in[8];
    const float* Wr3  = (const float*)d_in[9];
    const float* Wn3  = (const float*)d_in[10];
    const float* b3   = (const float*)d_in[11];
    const float* p1   = (const float*)d_in[12];
    const float* p2   = (const float*)d_in[13];
    const float* p3   = (const float*)d_in[14];
    const float* Wl1  = (const float*)d_in[15];
    const float* bl1  = (const float*)d_in[16];
    const float* Wl3  = (const float*)d_in[17];
    const float* bl3  = (const float*)d_in[18];
    float* out = (float*)d_out;

    // workspace layout
    char* ws = (char*)d_ws;
    float* agg = (float*)ws;                 ws += (size_t)NT_ * F0_ * sizeof(float); // 64MB
    float* xa  = (float*)ws;                 ws += (size_t)NT_ * H_  * sizeof(float); // 32MB
    float* xb  = (float*)ws;                 ws += (size_t)NT_ * H_  * sizeof(float); // 32MB
    float* r   = (float*)ws;                 ws += (size_t)B_ * 64 * sizeof(float);
    unsigned char* mask = (unsigned char*)ws;

    hipMemsetAsync(r, 0, (size_t)B_ * 64 * sizeof(float), stream);
    hipMemsetAsync(mask, 1, (size_t)NT_, stream);   // all nodes active

    // ---- stage 1 (F0=64 -> H=32) ----
    agg_kernel<F0_><<<dim3(B_, F0_ / 8), 512, 0, stream>>>(x, esrc, edst, agg);
    conv_kernel<F0_><<<NT_ / 128, 256, 0, stream>>>(x, agg, Wr1, Wn1, b1, xa);
    pool_kernel<<<B_, 256, 0, stream>>>(xa, mask, p1, K1_, r);

    // ---- stage 2 (H=32) ----
    agg_kernel<H_><<<dim3(B_, H_ / 8), 512, 0, stream>>>(xa, esrc, edst, agg);
    conv_kernel<H_><<<NT_ / 128, 256, 0, stream>>>(xa, agg, Wr2, Wn2, b2, xb);
    pool_kernel<<<B_, 256, 0, stream>>>(xb, mask, p2, K2_, r);

    // ---- stage 3 (H=32) ----
    agg_kernel<H_><<<dim3(B_, H_ / 8), 512, 0, stream>>>(xb, esrc, edst, agg);
    conv_kernel<H_><<<NT_ / 128, 256, 0, stream>>>(xb, agg, Wr3, Wn3, b3, xa);
    pool_kernel<<<B_, 256, 0, stream>>>(xa, mask, p3, K3_, r);

    // ---- head ----
    head_kernel<<<1, 128, 0, stream>>>(r, Wl1, bl1, Wl3, bl3, out);
}